// StateUpdater_58995670778145
// MI455X (gfx1250) — compile-verified
//
#include <hip/hip_runtime.h>

typedef __attribute__((ext_vector_type(2))) float v2f;
typedef __attribute__((ext_vector_type(4))) float v4f;
typedef __attribute__((ext_vector_type(8))) float v8f;

// ---------------------------------------------------------------------------
// Kernel 1: copy state_table -> out_table (the scatter in kernel 2 then
// overwrites the touched rows). float4 grid-stride; 1M floats = 4 MB.
// ---------------------------------------------------------------------------
__global__ void su_copy_table(const float* __restrict__ src,
                              float* __restrict__ dst, int n4, int n) {
    int i = blockIdx.x * blockDim.x + threadIdx.x;
    int stride = gridDim.x * blockDim.x;
    const v4f* s4 = (const v4f*)src;
    v4f* d4 = (v4f*)dst;
    for (int k = i; k < n4; k += stride) d4[k] = s4[k];
    for (int k = 4 * n4 + i; k < n; k += stride) dst[k] = src[k];  // tail
}

// ---------------------------------------------------------------------------
// Kernel 2: per 16-node tile (one wave32 each):
//   dot(mem_row, W) via 32 x V_WMMA_F32_16X16X4_F32 (full fp32, RNE)
//   K remap: lanes 0-15 consume k in [0,64), lanes 16-31 k in [64,128)
//   so each lane streams half the row with contiguous b128 loads.
//   B is W broadcast across all 16 columns -> every D column holds the dot.
// ---------------------------------------------------------------------------
__global__ void su_update(const int* __restrict__ ids,
                          const float* __restrict__ state,
                          const float* __restrict__ mem,
                          const float* __restrict__ W,
                          const float* __restrict__ bvec,
                          float* __restrict__ out_table,
                          float* __restrict__ out_flags,
                          int n_ids) {
    const int lane = threadIdx.x & 31;
    const int wave = threadIdx.x >> 5;
    const int tile = blockIdx.x * (blockDim.x >> 5) + wave;
    const int base = tile * 16;
    if (base >= n_ids) return;  // wave-uniform: EXEC stays all-1s for WMMA

    const int m    = lane & 15;   // matrix row this lane feeds (A layout)
    const int half = lane >> 4;   // 0: K 0..63, 1: K 64..127

    int ridx = base + m;
    if (ridx >= n_ids) ridx = n_ids - 1;  // clamp dup for ragged tail (unused)
    const int id = ids[ridx];
    const float* rowp = mem + (size_t)id * 128 + half * 64;
    const float* wp   = W + half * 64;

    v8f acc = {0.f, 0.f, 0.f, 0.f, 0.f, 0.f, 0.f, 0.f};
#pragma unroll
    for (int j = 0; j < 16; ++j) {
        v4f a4 = *(const v4f*)(rowp + 4 * j);   // 16B-aligned (row base 512B)
        v4f w4 = *(const v4f*)(wp + 4 * j);
        v2f a0 = {a4.x, a4.y};
        v2f b0 = {w4.x, w4.y};
        acc = __builtin_amdgcn_wmma_f32_16x16x4_f32(
            false, a0, false, b0, (short)0, acc, false, false);
        v2f a1 = {a4.z, a4.w};
        v2f b1 = {w4.z, w4.w};
        acc = __builtin_amdgcn_wmma_f32_16x16x4_f32(
            false, a1, false, b1, (short)0, acc, false, false);
    }

    // D layout (16x16 f32): VGPR r holds row r (lanes 0-15) / row r+8
    // (lanes 16-31); all 16 columns identical. Lane picks its row via
    // a cndmask chain; lanes 0-7 finish rows 0-7, lanes 16-23 rows 8-15.
    const int j = lane & 7;
    float dot = acc[0];
    dot = (j == 1) ? acc[1] : dot;
    dot = (j == 2) ? acc[2] : dot;
    dot = (j == 3) ? acc[3] : dot;
    dot = (j == 4) ? acc[4] : dot;
    dot = (j == 5) ? acc[5] : dot;
    dot = (j == 6) ? acc[6] : dot;
    dot = (j == 7) ? acc[7] : dot;

    const bool active = (lane & 15) < 8;  // post-WMMA predication only
    const int row = j + half * 8;
    const int gid = base + row;
    if (active && gid < n_ids) {
        const int nid   = ids[gid];
        const float st  = state[nid];
        const float bias = bvec[0];
        const float dlt = 1.0f / (1.0f + expf(-(dot + bias)));   // sigmoid
        const float isu = (st > 0.0f) ? 1.0f : 0.0f;             // hard gate
        const float alt = st + fminf(dlt, 1.0f - st);
        const float ns  = isu * dlt + (1.0f - isu) * alt;
        out_table[nid] = ns;   // scatter (ids unique per reference)
        out_flags[gid] = isu;  // second tuple output
    }
}

extern "C" void kernel_launch(void* const* d_in, const int* in_sizes, int n_in,
                              void* d_out, int out_size, void* d_ws, size_t ws_size,
                              hipStream_t stream) {
    const int*   ids   = (const int*)d_in[0];    // [n_ids] int32
    const float* state = (const float*)d_in[1];  // [N_NODES, 1] f32
    const float* mem   = (const float*)d_in[2];  // [N_NODES, 128] f32
    const float* W     = (const float*)d_in[3];  // [1, 128] f32
    const float* b     = (const float*)d_in[4];  // [1] f32

    const int n_ids   = in_sizes[0];
    const int n_nodes = in_sizes[1];  // state_table flat count (D=1)

    float* out_table = (float*)d_out;            // [n_nodes] updated table
    float* out_flags = out_table + n_nodes;      // [n_ids] is_updated

    // 1) copy state_table into the output table
    int n4 = n_nodes >> 2;
    int cb = 256;
    int cg = (n4 + cb - 1) / cb;
    if (cg > 4096) cg = 4096;
    if (cg < 1) cg = 1;
    su_copy_table<<<cg, cb, 0, stream>>>(state, out_table, n4, n_nodes);

    // 2) gather + WMMA dot + gate + scatter (ordered after copy on stream)
    const int tiles = (n_ids + 15) >> 4;   // one wave32 per 16-node tile
    const int wavesPerBlock = 4;           // 128 threads/block
    const int ug = (tiles + wavesPerBlock - 1) / wavesPerBlock;
    su_update<<<ug, wavesPerBlock * 32, 0, stream>>>(
        ids, state, mem, W, b, out_table, out_flags, n_ids);
}